// MultiPositivePrototypeHead_72971494359567
// MI455X (gfx1250) — compile-verified
//
#include <hip/hip_runtime.h>
#include <math.h>

typedef float v2f __attribute__((ext_vector_type(2)));
typedef float v8f __attribute__((ext_vector_type(8)));
typedef int   v4i __attribute__((ext_vector_type(4)));

#define NTOT  131072
#define DIM   512
#define HDIM  128
#define KP    3
#define NPOS  8192
#define NNEG  8192

#if __has_builtin(__builtin_amdgcn_global_load_async_to_lds_b128)
#define USE_ASYNC 1
typedef __attribute__((address_space(1))) v4i gv4i;
typedef __attribute__((address_space(3))) v4i lv4i;
#else
#define USE_ASYNC 0
#endif

__device__ __forceinline__ void async_cp16(const float* g, float* l) {
#if USE_ASYNC
    __builtin_amdgcn_global_load_async_to_lds_b128((gv4i*)(g), (lv4i*)(l), 0, 0);
#else
    float4 v = *(const float4*)g;
    l[0] = v.x; l[1] = v.y; l[2] = v.z; l[3] = v.w;
#endif
}

__device__ __forceinline__ void wait_async0() {
#if USE_ASYNC
#if __has_builtin(__builtin_amdgcn_s_wait_asynccnt)
    __builtin_amdgcn_s_wait_asynccnt(0);
#else
    asm volatile("s_wait_asynccnt 0x0" ::: "memory");
#endif
#endif
}

// ---------------- workspace layout (floats) ----------------
// centers : [0,1536)    kmeans centers (raw)
// negp    : [1536,2048) final calibrated neg prototype
// posp    : [2048,3584) final calibrated pos prototypes
// cnorms  : [3584,3588) |posp0..2|^2, |negp|^2
// calibP  : [3588,4100) 0.15*(pos_ctx@Wp^T + bp)
// negacc  : [4100,4612) negative gather sum
// gsum    : [4612,6148) kmeans segment sums
// gcount  : [6148,6151) kmeans counts
// cnraw   : [6151,6154) |center_k|^2 (raw, per iteration)

__global__ void zero_k(float* p, int n) {
    int i = blockIdx.x * blockDim.x + threadIdx.x;
    if (i < n) p[i] = 0.f;
}

// Sum z_proto[neg_idx[r]] over 8192 rows into negacc[512].
__global__ __launch_bounds__(256) void neg_sum_k(const float* __restrict__ zp,
                                                 const int* __restrict__ nidx,
                                                 float* __restrict__ negacc) {
    __shared__ float nacc[DIM];
    int tid = threadIdx.x;
    for (int i = tid; i < DIM; i += 256) nacc[i] = 0.f;
    __syncthreads();
    int wave = tid >> 5, lane = tid & 31;
    float4 a[4];
    for (int c = 0; c < 4; ++c) { a[c].x = a[c].y = a[c].z = a[c].w = 0.f; }
    for (int i = 0; i < 16; ++i) {
        int r = blockIdx.x * 128 + wave * 16 + i;
        const float* x = zp + (long)nidx[r] * DIM;
        for (int c = 0; c < 4; ++c) {
            float4 v = *(const float4*)(x + c * 128 + lane * 4);
            a[c].x += v.x; a[c].y += v.y; a[c].z += v.z; a[c].w += v.w;
        }
    }
    for (int c = 0; c < 4; ++c) {
        int d = c * 128 + lane * 4;
        atomicAdd(&nacc[d + 0], a[c].x);
        atomicAdd(&nacc[d + 1], a[c].y);
        atomicAdd(&nacc[d + 2], a[c].z);
        atomicAdd(&nacc[d + 3], a[c].w);
    }
    __syncthreads();
    for (int i = tid; i < DIM; i += 256) atomicAdd(&negacc[i], nacc[i]);
}

// outv[j] = mode ? acc[j]/NNEG + 0.15*(v.M[j]+bias[j]) : 0.15*(v.M[j]+bias[j])
__global__ __launch_bounds__(128) void matvec_k(const float* __restrict__ v,
                                                const float* __restrict__ M,
                                                const float* __restrict__ bias,
                                                const float* __restrict__ acc,
                                                float* __restrict__ outv, int mode) {
    int j = blockIdx.x, tid = threadIdx.x;
    const float* row = M + (long)j * DIM;
    float p = 0.f;
    for (int d = tid; d < DIM; d += 128) p += v[d] * row[d];
    for (int off = 16; off >= 1; off >>= 1) p += __shfl_xor(p, off, 32);
    __shared__ float r[4];
    if ((tid & 31) == 0) r[tid >> 5] = p;
    __syncthreads();
    if (tid == 0) {
        float dot = r[0] + r[1] + r[2] + r[3];
        float calib = 0.15f * (dot + bias[j]);
        outv[j] = mode ? acc[j] * (1.f / (float)NNEG) + calib : calib;
    }
}

__global__ void km_init_k(const float* __restrict__ zp, const int* __restrict__ pidx,
                          float* __restrict__ centers, int s0, int s1, int s2) {
    int d = blockIdx.x * 256 + threadIdx.x;
    if (d >= DIM) return;
    int sel[3] = {s0, s1, s2};
    for (int k = 0; k < KP; ++k)
        centers[k * DIM + d] = zp[(long)pidx[sel[k]] * DIM + d];
}

// squared norms of vectors of length 512
__global__ __launch_bounds__(256) void vnorm_k(const float* __restrict__ v, float* __restrict__ outn) {
    int b = blockIdx.x, tid = threadIdx.x;
    __shared__ float red[256];
    float s = 0.f;
    for (int d = tid; d < DIM; d += 256) { float x = v[b * DIM + d]; s += x * x; }
    red[tid] = s; __syncthreads();
    for (int st = 128; st > 0; st >>= 1) { if (tid < st) red[tid] += red[tid + st]; __syncthreads(); }
    if (tid == 0) outn[b] = red[0];
}

// one kmeans iteration: assignment + LDS-accumulated segment sums -> global atomics
__global__ __launch_bounds__(256) void km_assign_k(const float* __restrict__ zp,
                                                   const int* __restrict__ pidx,
                                                   const float* __restrict__ centers,
                                                   const float* __restrict__ cn,
                                                   float* __restrict__ gsum,
                                                   float* __restrict__ gcount) {
    __shared__ float cs[KP * DIM];
    __shared__ float lsum[KP * DIM];
    __shared__ float lcnt[KP];
    int tid = threadIdx.x;
    for (int i = tid; i < KP * DIM; i += 256) { cs[i] = centers[i]; lsum[i] = 0.f; }
    if (tid < KP) lcnt[tid] = 0.f;
    __syncthreads();
    float c0n = cn[0], c1n = cn[1], c2n = cn[2];
    int wave = tid >> 5, lane = tid & 31;
    const float* cr0 = &cs[0];
    const float* cr1 = &cs[DIM];
    const float* cr2 = &cs[2 * DIM];
    for (int i = 0; i < 16; ++i) {
        int r = blockIdx.x * 128 + wave * 16 + i;
        const float* x = zp + (long)pidx[r] * DIM;
        float4 xv[4];
        float d0 = 0.f, d1 = 0.f, d2 = 0.f;
        for (int c = 0; c < 4; ++c) {
            int db = c * 128 + lane * 4;
            xv[c] = *(const float4*)(x + db);
            d0 += xv[c].x * cr0[db] + xv[c].y * cr0[db + 1] + xv[c].z * cr0[db + 2] + xv[c].w * cr0[db + 3];
            d1 += xv[c].x * cr1[db] + xv[c].y * cr1[db + 1] + xv[c].z * cr1[db + 2] + xv[c].w * cr1[db + 3];
            d2 += xv[c].x * cr2[db] + xv[c].y * cr2[db + 1] + xv[c].z * cr2[db + 2] + xv[c].w * cr2[db + 3];
        }
        for (int off = 16; off >= 1; off >>= 1) {
            d0 += __shfl_xor(d0, off, 32);
            d1 += __shfl_xor(d1, off, 32);
            d2 += __shfl_xor(d2, off, 32);
        }
        // argmin of |c|^2 - 2 x.c  (|x|^2 constant; first-index tie-break)
        float s0 = c0n - 2.f * d0, s1 = c1n - 2.f * d1, s2 = c2n - 2.f * d2;
        int k = 0; float best = s0;
        if (s1 < best) { best = s1; k = 1; }
        if (s2 < best) { best = s2; k = 2; }
        for (int c = 0; c < 4; ++c) {
            int db = k * DIM + c * 128 + lane * 4;
            atomicAdd(&lsum[db + 0], xv[c].x);
            atomicAdd(&lsum[db + 1], xv[c].y);
            atomicAdd(&lsum[db + 2], xv[c].z);
            atomicAdd(&lsum[db + 3], xv[c].w);
        }
        if (lane == 0) atomicAdd(&lcnt[k], 1.f);
    }
    __syncthreads();
    for (int i = tid; i < KP * DIM; i += 256) atomicAdd(&gsum[i], lsum[i]);
    if (tid < KP) atomicAdd(&gcount[tid], lcnt[tid]);
}

// centers update (empty cluster keeps old) + new squared norms
__global__ __launch_bounds__(256) void km_update_k(const float* __restrict__ gsum,
                                                   const float* __restrict__ gcount,
                                                   float* __restrict__ centers,
                                                   float* __restrict__ cn) {
    int k = blockIdx.x, tid = threadIdx.x;
    float cnt = gcount[k];
    __shared__ float red[256];
    float nsq = 0.f;
    for (int d = tid; d < DIM; d += 256) {
        float c = centers[k * DIM + d];
        if (cnt > 0.f) c = gsum[k * DIM + d] / fmaxf(cnt, 1.f);
        centers[k * DIM + d] = c;
        nsq += c * c;
    }
    red[tid] = nsq; __syncthreads();
    for (int s = 128; s > 0; s >>= 1) { if (tid < s) red[tid] += red[tid + s]; __syncthreads(); }
    if (tid == 0) cn[k] = red[0];
}

// pos_proto = centers + calib (broadcast); write prototypes to d_out; norms
__global__ __launch_bounds__(256) void calib_finalize_k(const float* __restrict__ centers,
                                                        const float* __restrict__ calibP,
                                                        const float* __restrict__ negp,
                                                        float* __restrict__ posp,
                                                        float* __restrict__ cnorms,
                                                        float* __restrict__ outPP) {
    int b = blockIdx.x, tid = threadIdx.x;
    __shared__ float red[256];
    float nsq = 0.f;
    if (b < KP) {
        for (int d = tid; d < DIM; d += 256) {
            float v = centers[b * DIM + d] + calibP[d];
            posp[b * DIM + d] = v;
            outPP[b * DIM + d] = v;
            nsq += v * v;
        }
    } else {
        for (int d = tid; d < DIM; d += 256) {
            float v = negp[d];
            outPP[KP * DIM + d] = v;
            nsq += v * v;
        }
    }
    red[tid] = nsq; __syncthreads();
    for (int s = 128; s > 0; s >>= 1) { if (tid < s) red[tid] += red[tid + s]; __syncthreads(); }
    if (tid == 0) cnorms[b] = red[0];
}

// ---------------- fused main kernel ----------------
// 64 rows per block, 4 waves. Phase 1: h = relu(z_cls@W1^T+b1) via WMMA f32 16x16x4
// with async-to-LDS double-buffered staging. Phase 2: prototype distances.
#define LDK 36  // padded LDS stride for 32-float K-chunk rows

__global__ __launch_bounds__(128) void fused_main_k(
    const float* __restrict__ zc, const float* __restrict__ zp,
    const float* __restrict__ W1, const float* __restrict__ b1,
    const float* __restrict__ W2, const float* __restrict__ b2,
    const float* __restrict__ posp, const float* __restrict__ negp,
    const float* __restrict__ cnorms, float* __restrict__ out) {
    __shared__ float As[2][64 * LDK];
    __shared__ float Bs[2][HDIM * LDK];
    __shared__ float protos[4 * DIM];
    __shared__ float b1s[HDIM];
    __shared__ float w2s[HDIM];
    __shared__ float clsl[64];
    __shared__ float cns[4];

    int tid = threadIdx.x;
    int wave = tid >> 5, lane = tid & 31;
    int m = lane & 15, hi = lane >> 4;
    long row0 = (long)blockIdx.x * 64;

    for (int i = tid; i < KP * DIM; i += 128) protos[i] = posp[i];
    for (int i = tid; i < DIM; i += 128) protos[KP * DIM + i] = negp[i];
    if (tid < HDIM) { b1s[tid] = b1[tid]; w2s[tid] = W2[tid]; }
    if (tid < 4) cns[tid] = cnorms[tid];

    v8f acc[8];
    for (int t = 0; t < 8; ++t)
        for (int e = 0; e < 8; ++e) acc[t][e] = 0.f;

    // A tile: 64 rows x 32k (512 float4s); B tile: 128 n x 32k (1024 float4s)
#define STAGE_CHUNK(kc, buf)                                                    \
    {                                                                           \
        int kbase = (kc) * 32;                                                  \
        for (int q = 0; q < 4; ++q) {                                           \
            int id = q * 128 + tid;                                             \
            int r = id >> 3, kq = (id & 7) * 4;                                 \
            async_cp16(zc + (row0 + r) * DIM + kbase + kq,                      \
                       &As[buf][r * LDK + kq]);                                 \
        }                                                                       \
        for (int q = 0; q < 8; ++q) {                                           \
            int id = q * 128 + tid;                                             \
            int n = id >> 3, kq = (id & 7) * 4;                                 \
            async_cp16(W1 + (long)n * DIM + kbase + kq,                         \
                       &Bs[buf][n * LDK + kq]);                                 \
        }                                                                       \
    }

    STAGE_CHUNK(0, 0);
    wait_async0();
    __syncthreads();

    int arow = (wave * 16 + m) * LDK;
    for (int kc = 0; kc < DIM / 32; ++kc) {
        int cur = kc & 1;
        if (kc + 1 < DIM / 32) STAGE_CHUNK(kc + 1, 1 - cur);
        const float* Ac = &As[cur][0];
        const float* Bc = &Bs[cur][0];
        // 4 k-step pairs per chunk; prefetch all fragments, then 16-WMMA chain
        for (int sp = 0; sp < 4; ++sp) {
            int kk = sp * 8 + hi * 2;
            v2f a0, a1;
            a0.x = Ac[arow + kk];     a0.y = Ac[arow + kk + 1];
            a1.x = Ac[arow + kk + 4]; a1.y = Ac[arow + kk + 5];
            v2f b0[8], b1f[8];
#pragma unroll
            for (int t = 0; t < 8; ++t) {
                int brow = (t * 16 + m) * LDK;
                b0[t].x  = Bc[brow + kk];     b0[t].y  = Bc[brow + kk + 1];
                b1f[t].x = Bc[brow + kk + 4]; b1f[t].y = Bc[brow + kk + 5];
            }
#pragma unroll
            for (int t = 0; t < 8; ++t)
                acc[t] = __builtin_amdgcn_wmma_f32_16x16x4_f32(
                    false, a0, false, b0[t], (short)0, acc[t], false, false);
#pragma unroll
            for (int t = 0; t < 8; ++t)
                acc[t] = __builtin_amdgcn_wmma_f32_16x16x4_f32(
                    false, a1, false, b1f[t], (short)0, acc[t], false, false);
        }
        wait_async0();
        __syncthreads();
    }

    // fused bias + relu + W2 reduction. C layout: VGPR v holds rows (v + hi*8), N = m.
    float b2v = b2[0];
    float partial[8];
    for (int v = 0; v < 8; ++v) partial[v] = 0.f;
    for (int t = 0; t < 8; ++t) {
        float w2v = w2s[t * 16 + m];
        float b1v = b1s[t * 16 + m];
        for (int v = 0; v < 8; ++v) {
            float h = fmaxf(acc[t][v] + b1v, 0.f);
            partial[v] += h * w2v;
        }
    }
    for (int v = 0; v < 8; ++v) {
        float p = partial[v];
        p += __shfl_xor(p, 1, 32);
        p += __shfl_xor(p, 2, 32);
        p += __shfl_xor(p, 4, 32);
        p += __shfl_xor(p, 8, 32);
        if (m == 0) clsl[wave * 16 + hi * 8 + v] = p + b2v;
    }
    __syncthreads();

    // Phase 2: two threads per row, 256 dims each
    int row = tid >> 1, half = tid & 1;
    const float* zrow = zp + (row0 + row) * DIM + half * 256;
    float zz = 0.f, d0 = 0.f, d1 = 0.f, d2 = 0.f, dn = 0.f;
    for (int j = 0; j < 64; ++j) {
        float4 z4 = *(const float4*)(zrow + j * 4);
        int db = half * 256 + j * 4;
        const float* c0 = &protos[db];
        const float* c1 = &protos[DIM + db];
        const float* c2 = &protos[2 * DIM + db];
        const float* cg = &protos[3 * DIM + db];
        zz += z4.x * z4.x + z4.y * z4.y + z4.z * z4.z + z4.w * z4.w;
        d0 += z4.x * c0[0] + z4.y * c0[1] + z4.z * c0[2] + z4.w * c0[3];
        d1 += z4.x * c1[0] + z4.y * c1[1] + z4.z * c1[2] + z4.w * c1[3];
        d2 += z4.x * c2[0] + z4.y * c2[1] + z4.z * c2[2] + z4.w * c2[3];
        dn += z4.x * cg[0] + z4.y * cg[1] + z4.z * cg[2] + z4.w * cg[3];
    }
    zz += __shfl_xor(zz, 1, 32);
    d0 += __shfl_xor(d0, 1, 32);
    d1 += __shfl_xor(d1, 1, 32);
    d2 += __shfl_xor(d2, 1, 32);
    dn += __shfl_xor(dn, 1, 32);
    if (half == 0) {
        float dp0 = fmaxf(zz + cns[0] - 2.f * d0, 0.f);
        float dp1 = fmaxf(zz + cns[1] - 2.f * d1, 0.f);
        float dp2 = fmaxf(zz + cns[2] - 2.f * d2, 0.f);
        float dmin = dp0; int bi = 0;
        if (dp1 < dmin) { dmin = dp1; bi = 1; }
        if (dp2 < dmin) { dmin = dp2; bi = 2; }
        float dneg = (zz - 2.f * dn + cns[3]) * (1.f / (float)DIM);
        float plog = (dneg - dmin) * 0.04419417382415922f; // 1/sqrt(512)
        float pterm = 0.2f * plog;
        float cls = clsl[row];
        float logit = (cls + pterm) * (1.f / 2.5f);
        float prob = 1.f / (1.f + expf(-logit));
        long gn = row0 + row;
        out[gn] = logit;
        out[(long)NTOT + gn] = prob;
        out[2L * NTOT + gn] = pterm;
        out[3L * NTOT + gn] = (float)bi;
        out[4L * NTOT + gn * 3 + 0] = dp0;
        out[4L * NTOT + gn * 3 + 1] = dp1;
        out[4L * NTOT + gn * 3 + 2] = dp2;
    }
}

extern "C" void kernel_launch(void* const* d_in, const int* in_sizes, int n_in,
                              void* d_out, int out_size, void* d_ws, size_t ws_size,
                              hipStream_t stream) {
    const float* z_cls   = (const float*)d_in[0];
    const float* z_proto = (const float*)d_in[1];
    const float* pos_ctx = (const float*)d_in[2];
    const float* neg_ctx = (const float*)d_in[3];
    const float* W1 = (const float*)d_in[4];
    const float* b1 = (const float*)d_in[5];
    const float* W2 = (const float*)d_in[6];
    const float* b2 = (const float*)d_in[7];
    const float* Wp = (const float*)d_in[8];
    const float* bp = (const float*)d_in[9];
    const float* Wn = (const float*)d_in[10];
    const float* bn = (const float*)d_in[11];
    const int* pos_idx = (const int*)d_in[12];
    const int* neg_idx = (const int*)d_in[13];
    float* out = (float*)d_out;

    float* wsf     = (float*)d_ws;
    float* centers = wsf;           // 1536
    float* negp    = wsf + 1536;    // 512
    float* posp    = wsf + 2048;    // 1536
    float* cnorms  = wsf + 3584;    // 4
    float* calibP  = wsf + 3588;    // 512
    float* negacc  = wsf + 4100;    // 512
    float* gsum    = wsf + 4612;    // 1536
    float* gcount  = wsf + 6148;    // 3
    float* cnraw   = wsf + 6151;    // 3

    // negative prototype: mean + calibration
    zero_k<<<2, 256, 0, stream>>>(negacc, DIM);
    neg_sum_k<<<64, 256, 0, stream>>>(z_proto, neg_idx, negacc);
    matvec_k<<<DIM, 128, 0, stream>>>(neg_ctx, Wn, bn, negacc, negp, 1);

    // kmeans init: linspace(0, NPOS-1, K) truncated -> {0, 4095, 8191}
    int s0 = 0;
    int s1 = (int)((float)(NPOS - 1) * 1.f / (float)(KP - 1));
    int s2 = NPOS - 1;
    km_init_k<<<2, 256, 0, stream>>>(z_proto, pos_idx, centers, s0, s1, s2);
    vnorm_k<<<KP, 256, 0, stream>>>(centers, cnraw);

    for (int it = 0; it < 4; ++it) {
        zero_k<<<7, 256, 0, stream>>>(gsum, KP * DIM + KP); // sums + counts contiguous
        km_assign_k<<<64, 256, 0, stream>>>(z_proto, pos_idx, centers, cnraw, gsum, gcount);
        km_update_k<<<KP, 256, 0, stream>>>(gsum, gcount, centers, cnraw);
    }

    // positive calibration + finalize prototypes (also writes prototype outputs)
    matvec_k<<<DIM, 128, 0, stream>>>(pos_ctx, Wp, bp, negacc, calibP, 0);
    calib_finalize_k<<<KP + 1, 256, 0, stream>>>(centers, calibP, negp, posp, cnorms,
                                                 out + 7L * NTOT);

    // fused main pass: WMMA MLP + prototype distances, one read of z_cls/z_proto
    fused_main_k<<<NTOT / 64, 128, 0, stream>>>(z_cls, z_proto, W1, b1, W2, b2,
                                                posp, negp, cnorms, out);
}